// SWM_85349590106443
// MI455X (gfx1250) — compile-verified
//
#include <hip/hip_runtime.h>

#define NX 2048
#define NY 2048
#define TS 32

#define GRAV   9.81f
#define F0C    2e-4f
#define BETAC  2e-11f
#define INV_DX (1.0f / 5.0e3f)
#define INV_DY (1.0f / 5.0e3f)
#define WEPS   1e-8f

// LDS tile strides (odd to dodge bank conflicts)
#define HS 39   // h tile: 38 rows x 38 cols, stride 39
#define US 35   // u/v tiles: 34 x 34, stride 35
#define QS 33   // intermediates: 33 x 33, stride 33

__device__ __forceinline__ int clampi(int x, int lo, int hi) {
    return x < lo ? lo : (x > hi ? hi : x);
}

// CDNA5 async global->LDS copy (GVS mode: saddr base + 32-bit vgpr byte offset).
// Tracked by ASYNCcnt; completed via s_wait_asynccnt.
__device__ __forceinline__ void async_copy_b32(unsigned lds_byte, const float* base,
                                               unsigned glb_byte) {
    asm volatile("global_load_async_to_lds_b32 %0, %1, %2"
                 :
                 : "v"(lds_byte), "v"(glb_byte), "s"(base)
                 : "memory");
}

__device__ __forceinline__ void wait_async_all() {
    asm volatile("s_wait_asynccnt 0x0" ::: "memory");
}

__device__ __forceinline__ unsigned lds_off(const void* p) {
    // low 32 bits of the flat shared-aperture address == LDS byte offset
    return (unsigned)(unsigned long long)p;
}

// WENO-Z 5th order at the right interface of cell f2
__device__ __forceinline__ float wenoz5(float f0, float f1, float f2, float f3, float f4) {
    const float c1312 = 13.0f / 12.0f;
    float d0 = f0 - 2.0f * f1 + f2;
    float d1 = f1 - 2.0f * f2 + f3;
    float d2 = f2 - 2.0f * f3 + f4;
    float e0 = f0 - 4.0f * f1 + 3.0f * f2;
    float e1 = f1 - f3;
    float e2 = 3.0f * f2 - 4.0f * f3 + f4;
    float b0 = c1312 * d0 * d0 + 0.25f * e0 * e0;
    float b1 = c1312 * d1 * d1 + 0.25f * e1 * e1;
    float b2 = c1312 * d2 * d2 + 0.25f * e2 * e2;
    float tau = fabsf(b0 - b2);
    float r0 = tau / (b0 + WEPS);
    float r1 = tau / (b1 + WEPS);
    float r2 = tau / (b2 + WEPS);
    float a0 = 0.1f * (1.0f + r0 * r0);
    float a1 = 0.6f * (1.0f + r1 * r1);
    float a2 = 0.3f * (1.0f + r2 * r2);
    const float s = 1.0f / 6.0f;
    float p0 = (2.0f * f0 - 7.0f * f1 + 11.0f * f2) * s;
    float p1 = (-f1 + 5.0f * f2 + 2.0f * f3) * s;
    float p2 = (2.0f * f2 + 5.0f * f3 - f4) * s;
    return (a0 * p0 + a1 * p1 + a2 * p2) / (a0 + a1 + a2);
}

// Upwind-selected WENO: equivalent to (up ? weno(f0..f4) : weno(f5,f4,f3,f2,f1)),
// but selects the 5 inputs first so only ONE WENO evaluation (and its 4 divides)
// is needed instead of two.
__device__ __forceinline__ float weno_upwind(bool up, float f0, float f1, float f2,
                                             float f3, float f4, float f5) {
    float a0 = up ? f0 : f5;
    float a1 = up ? f1 : f4;
    float a2 = up ? f2 : f3;
    float a3 = up ? f3 : f2;
    float a4 = up ? f4 : f1;
    return wenoz5(a0, a1, a2, a3, a4);
}

__global__ __launch_bounds__(256) void SWM_85349590106443_kernel(
    const float* __restrict__ h, const float* __restrict__ u, const float* __restrict__ v,
    const float* __restrict__ mc, const float* __restrict__ mu, const float* __restrict__ mv,
    const float* __restrict__ Yq,
    float* __restrict__ out_h, float* __restrict__ out_u, float* __restrict__ out_v) {

    __shared__ float sh_h[38 * HS];   // h cells [i0-3 .. i0+34] x [j0-3 .. j0+34]
    __shared__ float sh_u[34 * US];   // u faces k in [i0-1..i0+32], cells j in [j0-1..j0+32]
    __shared__ float sh_v[34 * US];   // v cells i in [i0-1..i0+32], faces l in [j0-1..j0+32]
    __shared__ float sh_uh[33 * QS];  // uh: faces k in [i0..i0+32], cells j in [j0-1..j0+31]
    __shared__ float sh_vh[33 * QS];  // vh: cells i in [i0-1..i0+31], faces l in [j0..j0+32]
    __shared__ float sh_p [33 * QS];  // p : cells [i0-1..i0+31] x [j0-1..j0+31]
    __shared__ float sh_q [33 * QS];  // q : corners [i0..i0+32] x [j0..j0+32]

    const int tid = threadIdx.x;
    const int i0 = blockIdx.y * TS;
    const int j0 = blockIdx.x * TS;

    // ---------------- Phase 0: async stage h, u, v into LDS ----------------
    {
        const unsigned bh = lds_off(sh_h);
        const unsigned bu = lds_off(sh_u);
        const unsigned bv = lds_off(sh_v);

        // h: 38x38 = 1444 elements, clamped (edge ghost cells)
        for (int t = 0; t < 6; ++t) {
            int idx = tid + t * 256; idx = idx < 1444 ? idx : 1443;
            int r = idx / 38, c = idx % 38;
            int gi = clampi(i0 - 3 + r, 0, NX - 1);
            int gj = clampi(j0 - 3 + c, 0, NY - 1);
            async_copy_b32(bh + (unsigned)((r * HS + c) * 4), h,
                           (unsigned)((gi * NY + gj) * 4));
        }
        // u: 34x34 = 1156 elements (faces x cells)
        for (int t = 0; t < 5; ++t) {
            int idx = tid + t * 256; idx = idx < 1156 ? idx : 1155;
            int r = idx / 34, c = idx % 34;
            int gk = clampi(i0 - 1 + r, 0, NX);
            int gj = clampi(j0 - 1 + c, 0, NY - 1);
            async_copy_b32(bu + (unsigned)((r * US + c) * 4), u,
                           (unsigned)((gk * NY + gj) * 4));
        }
        // v: 34x34 = 1156 elements (cells x faces)
        for (int t = 0; t < 5; ++t) {
            int idx = tid + t * 256; idx = idx < 1156 ? idx : 1155;
            int r = idx / 34, c = idx % 34;
            int gi = clampi(i0 - 1 + r, 0, NX - 1);
            int gl = clampi(j0 - 1 + c, 0, NY);
            async_copy_b32(bv + (unsigned)((r * US + c) * 4), v,
                           (unsigned)((gi * (NY + 1) + gl) * 4));
        }
        wait_async_all();
    }
    __syncthreads();

    // ---------------- Phase A: face fluxes, Bernoulli p, PV q ----------------
    for (int idx = tid; idx < 33 * 33; idx += 256) {
        const int a = idx / 33, b = idx % 33;

        // uh at face k = i0+a, cell j = j0-1+b  (x-direction WENO upwind)
        {
            float f0 = sh_h[(a + 0) * HS + (b + 2)];
            float f1 = sh_h[(a + 1) * HS + (b + 2)];
            float f2 = sh_h[(a + 2) * HS + (b + 2)];
            float f3 = sh_h[(a + 3) * HS + (b + 2)];
            float f4 = sh_h[(a + 4) * HS + (b + 2)];
            float f5 = sh_h[(a + 5) * HS + (b + 2)];
            float uu = sh_u[(a + 1) * US + b];
            float hsel = weno_upwind(uu >= 0.0f, f0, f1, f2, f3, f4, f5);
            int k = i0 + a;
            int jc = clampi(j0 - 1 + b, 0, NY - 1);
            sh_uh[a * QS + b] = uu * hsel * mu[k * NY + jc];
        }
        // vh at cell i = i0-1+a, face l = j0+b  (y-direction WENO upwind)
        {
            float g0 = sh_h[(a + 2) * HS + (b + 0)];
            float g1 = sh_h[(a + 2) * HS + (b + 1)];
            float g2 = sh_h[(a + 2) * HS + (b + 2)];
            float g3 = sh_h[(a + 2) * HS + (b + 3)];
            float g4 = sh_h[(a + 2) * HS + (b + 4)];
            float g5 = sh_h[(a + 2) * HS + (b + 5)];
            float vv = sh_v[a * US + (b + 1)];
            float hsel = weno_upwind(vv >= 0.0f, g0, g1, g2, g3, g4, g5);
            int ic = clampi(i0 - 1 + a, 0, NX - 1);
            int l = j0 + b;
            sh_vh[a * QS + b] = vv * hsel * mv[ic * (NY + 1) + l];
        }
        // p = g*h + ke at cell (i0-1+a, j0-1+b)
        {
            float uc = 0.5f * (sh_u[a * US + b] + sh_u[(a + 1) * US + b]);
            float vc = 0.5f * (sh_v[a * US + b] + sh_v[a * US + (b + 1)]);
            float hh = sh_h[(a + 2) * HS + (b + 2)];
            sh_p[a * QS + b] = GRAV * hh + 0.5f * (uc * uc + vc * vc);
        }
        // q at corner (c,d) = (i0+a, j0+b); boundary via clamp + sign flip
        {
            int c = i0 + a, d = j0 + b;
            int cc = clampi(c, 1, NX - 1);
            int dd = clampi(d, 1, NY - 1);
            int lc = cc - (i0 - 1), ld = dd - (j0 - 1);
            float dvdx = (sh_v[lc * US + ld] - sh_v[(lc - 1) * US + ld]) * INV_DX;
            float dudy = (sh_u[lc * US + ld] - sh_u[lc * US + (ld - 1)]) * INV_DY;
            int hrw = cc - (i0 - 3), hcl = dd - (j0 - 3);
            float hq = 0.25f * (sh_h[(hrw - 1) * HS + (hcl - 1)] + sh_h[hrw * HS + (hcl - 1)] +
                                sh_h[(hrw - 1) * HS + hcl]       + sh_h[hrw * HS + hcl]);
            float f = F0C + Yq[cc * (NY + 1) + dd] * BETAC;
            float qv = (dvdx - dudy + f) / hq;
            sh_q[a * QS + b] = (c == cc && d == dd) ? qv : -qv;
        }
    }
    __syncthreads();

    // ---------------- Phase B: assemble h_rhs, u_rhs, v_rhs ----------------
    for (int t = 0; t < 4; ++t) {
        int idx = tid + t * 256;          // 0..1023 over the 32x32 tile
        int ty = idx >> 5, tx = idx & 31;
        int i = i0 + ty, j = j0 + tx;

        float uhW = sh_uh[ty * QS + (tx + 1)];        // uh[i,   j]
        float uhE = sh_uh[(ty + 1) * QS + (tx + 1)];  // uh[i+1, j]
        float vhS = sh_vh[(ty + 1) * QS + tx];        // vh[i, j]
        float vhN = sh_vh[(ty + 1) * QS + (tx + 1)];  // vh[i, j+1]

        out_h[i * NY + j] =
            -((uhE - uhW) * INV_DX + (vhN - vhS) * INV_DY) * mc[i * NY + j];

        // u momentum at face k = i
        float ur = 0.0f;
        if (i != 0) {
            float vh00 = sh_vh[ty * QS + tx];
            float vh01 = sh_vh[ty * QS + (tx + 1)];
            float vhu = 0.25f * (vh00 + vh01 + vhS + vhN);
            float qu = 0.5f * (sh_q[ty * QS + tx] + sh_q[ty * QS + (tx + 1)]);
            float dp = (sh_p[(ty + 1) * QS + (tx + 1)] - sh_p[ty * QS + (tx + 1)]) * INV_DX;
            ur = (qu * vhu - dp) * mu[i * NY + j];
        }
        out_u[i * NY + j] = ur;

        // v momentum at face l = j
        float vr = 0.0f;
        if (j != 0) {
            float uh00 = sh_uh[ty * QS + tx];
            float uh10 = sh_uh[(ty + 1) * QS + tx];
            float uhv = 0.25f * (uh00 + uhW + uh10 + uhE);
            float qv_ = 0.5f * (sh_q[ty * QS + tx] + sh_q[(ty + 1) * QS + tx]);
            float dp = (sh_p[(ty + 1) * QS + (tx + 1)] - sh_p[(ty + 1) * QS + tx]) * INV_DY;
            vr = (-qv_ * uhv - dp) * mv[i * (NY + 1) + j];
        }
        out_v[i * (NY + 1) + j] = vr;
    }

    // boundary face rows/cols (jnp.pad zeros -> exactly zero)
    if (i0 + TS == NX && tid < TS) out_u[NX * NY + (j0 + tid)] = 0.0f;
    if (j0 + TS == NY && tid < TS) out_v[(i0 + tid) * (NY + 1) + NY] = 0.0f;
}

extern "C" void kernel_launch(void* const* d_in, const int* in_sizes, int n_in,
                              void* d_out, int out_size, void* d_ws, size_t ws_size,
                              hipStream_t stream) {
    (void)in_sizes; (void)n_in; (void)out_size; (void)d_ws; (void)ws_size;
    const float* h  = (const float*)d_in[0];
    const float* u  = (const float*)d_in[1];
    const float* v  = (const float*)d_in[2];
    const float* mc = (const float*)d_in[3];
    const float* mu = (const float*)d_in[4];
    const float* mv = (const float*)d_in[5];
    const float* Yq = (const float*)d_in[6];

    float* out   = (float*)d_out;
    float* out_h = out;                       // [2048, 2048]
    float* out_u = out_h + NX * NY;           // [2049, 2048]
    float* out_v = out_u + (NX + 1) * NY;     // [2048, 2049]

    dim3 grid(NY / TS, NX / TS);
    SWM_85349590106443_kernel<<<grid, 256, 0, stream>>>(h, u, v, mc, mu, mv, Yq,
                                                        out_h, out_u, out_v);
}